// drebin_42932493090827
// MI455X (gfx1250) — compile-verified
//
#include <hip/hip_runtime.h>
#include <math.h>

#define NQ     16
#define NBATCH 512
#define NCLS   100
#define NG     59      // gate-matrix table entries per sample

// d_ws layout in floats
#define WS_FEATS 0                              // 8192
#define WS_GTAB  8192                           // 512*59*8 = 241664
#define WS_XMAT  (8192 + NBATCH*NG*8)           // 2*512*16 = 16384
#define WS_XTRA  (WS_XMAT + 2*NBATCH*16)        // 16
#define WS_STATE (WS_XTRA + 16)                 // 512*65536*2 floats (256 MB), 16B aligned

struct cplx { float x, y; };
__device__ __forceinline__ cplx cmul(cplx a, cplx b){ return { a.x*b.x - a.y*b.y, a.x*b.y + a.y*b.x }; }
__device__ __forceinline__ cplx cadd(cplx a, cplx b){ return { a.x+b.x, a.y+b.y }; }

__device__ __forceinline__ void mix2(float& r0, float& i0, float& r1, float& i1,
                                     cplx u00, cplx u01, cplx u10, cplx u11) {
  float nr0 = u00.x*r0 - u00.y*i0 + u01.x*r1 - u01.y*i1;
  float ni0 = u00.x*i0 + u00.y*r0 + u01.x*i1 + u01.y*r1;
  float nr1 = u10.x*r0 - u10.y*i0 + u11.x*r1 - u11.y*i1;
  float ni1 = u10.x*i0 + u10.y*r0 + u11.x*i1 + u11.y*r1;
  r0 = nr0; i0 = ni0; r1 = nr1; i1 = ni1;
}
__device__ __forceinline__ void loadmat(const float* g, cplx* U) {
  U[0] = {g[0], g[1]}; U[1] = {g[2], g[3]}; U[2] = {g[4], g[5]}; U[3] = {g[6], g[7]};
}
__device__ __forceinline__ void mixp(float2& a, float2& b, const cplx* U) {
  mix2(a.x, a.y, b.x, b.y, U[0], U[1], U[2], U[3]);
}

// ---- CDNA5 async global<->LDS movers (ASYNCcnt-tracked, no VGPR staging) ----
__device__ __forceinline__ unsigned lds_off32(const void* p) {
  // generic pointer to __shared__ -> low 32 bits = wave-relative LDS byte address
  return (unsigned)(uintptr_t)p;
}
__device__ __forceinline__ void async_g2l_b128(const void* gptr, void* lptr) {
  asm volatile("global_load_async_to_lds_b128 %0, %1, off"
               :: "v"(lds_off32(lptr)), "v"((unsigned long long)(uintptr_t)gptr)
               : "memory");
}
__device__ __forceinline__ void async_l2g_b128(void* gptr, const void* lptr) {
  asm volatile("global_store_async_from_lds_b128 %0, %1, off"
               :: "v"((unsigned long long)(uintptr_t)gptr), "v"(lds_off32(lptr))
               : "memory");
}
__device__ __forceinline__ void wait_async0() {
  asm volatile("s_wait_asynccnt 0" ::: "memory");
}

// ---------------- prep: build per-sample gate-matrix tables ----------------
// gtab[s][g][8]: g=0..14 fused(RX,Rot) L1 wire g; 15..28 CRX L1 ring i=g-15;
//                g=29 RX(crx1[15]) (CRX(15,0) L1); 30..44 fused L2; 45..58 CRX L2.
// xmat[L][s][16]: M0 = F15 (bit14=0), M1 = RX(crx[14])*F15 (bit14=1)  (cross gate)
// xtra[0..1]: cos/sin of crx2[15]/2 for the final CRX(15,0)@L2.
__device__ __forceinline__ void rx_of(float th, cplx* U) {
  float c = cosf(0.5f*th), s = sinf(0.5f*th);
  U[0] = {c, 0.f}; U[1] = {0.f, -s}; U[2] = {0.f, -s}; U[3] = {c, 0.f};
}
__device__ __forceinline__ void fused_of(float xv, const float* rotw, cplx* F) {
  float cx = cosf(0.5f*xv), sx = sinf(0.5f*xv);
  cplx X0{cx, 0.f}, X1{0.f, -sx}, X2{0.f, -sx}, X3{cx, 0.f};
  float phi = rotw[0], tht = rotw[1], om = rotw[2];
  float c = cosf(0.5f*tht), sv = sinf(0.5f*tht);
  float al = 0.5f*(phi + om), be = 0.5f*(phi - om);
  float ca = cosf(al), sa = sinf(al), cb = cosf(be), sb = sinf(be);
  cplx R0{ ca*c, -sa*c }, R1{ -cb*sv, -sb*sv }, R2{ cb*sv, -sb*sv }, R3{ ca*c, sa*c };
  F[0] = cadd(cmul(R0, X0), cmul(R1, X2));
  F[1] = cadd(cmul(R0, X1), cmul(R1, X3));
  F[2] = cadd(cmul(R2, X0), cmul(R3, X2));
  F[3] = cadd(cmul(R2, X1), cmul(R3, X3));
}
__device__ __forceinline__ void matmul2(const cplx* A, const cplx* B, cplx* C) {
  C[0] = cadd(cmul(A[0], B[0]), cmul(A[1], B[2]));
  C[1] = cadd(cmul(A[0], B[1]), cmul(A[1], B[3]));
  C[2] = cadd(cmul(A[2], B[0]), cmul(A[3], B[2]));
  C[3] = cadd(cmul(A[2], B[1]), cmul(A[3], B[3]));
}
__device__ __forceinline__ void storemat(float* g, const cplx* M) {
  #pragma unroll
  for (int i = 0; i < 4; ++i) { g[2*i] = M[i].x; g[2*i+1] = M[i].y; }
}

__global__ __launch_bounds__(256) void prep_kernel(
    const float* __restrict__ x,    const float* __restrict__ rot1,
    const float* __restrict__ crx1, const float* __restrict__ rot2,
    const float* __restrict__ crx2, float* __restrict__ gtab,
    float* __restrict__ xmat,       float* __restrict__ xtra)
{
  int tid = blockIdx.x * 256 + threadIdx.x;   // 512 * 64
  int s = tid >> 6, q = tid & 63;
  if (s >= NBATCH) return;
  cplx M[4];
  if (q < 15)       { fused_of(x[s*NQ + q], rot1 + q*3, M);        storemat(gtab + ((size_t)s*NG + q)*8, M); }
  else if (q < 29)  { rx_of(crx1[q-15], M);                        storemat(gtab + ((size_t)s*NG + q)*8, M); }
  else if (q == 29) { rx_of(crx1[15], M);                          storemat(gtab + ((size_t)s*NG + q)*8, M); }
  else if (q < 45)  { fused_of(x[s*NQ + (q-30)], rot2 + (q-30)*3, M); storemat(gtab + ((size_t)s*NG + q)*8, M); }
  else if (q < 59)  { rx_of(crx2[q-45], M);                        storemat(gtab + ((size_t)s*NG + q)*8, M); }
  else if (q == 59 || q == 60) {
    int L = q - 59;
    cplx F[4]; fused_of(x[s*NQ + 15], (L ? rot2 : rot1) + 15*3, F);
    cplx X[4]; rx_of((L ? crx2 : crx1)[14], X);
    cplx M1[4]; matmul2(X, F, M1);
    float* xm = xmat + (size_t)(L*NBATCH + s)*16;
    storemat(xm, F); storemat(xm + 8, M1);
  } else if (q == 61 && s == 0) {
    xtra[0] = cosf(0.5f*crx2[15]);
    xtra[1] = sinf(0.5f*crx2[15]);
  }
}

// ---------------- local pass: half-sample (256 KB) fully resident in LDS ----------------
// block = (s, hb=bit15); phase 0: init |0..0> + L1 wires 0..14; phase 1: CRX(15,0)@L1 + L2 wires 0..14.
__global__ __launch_bounds__(1024) void local_pass(float2* __restrict__ state,
                                                   const float* __restrict__ gtab,
                                                   int phase)
{
  __shared__ __align__(16) float2 A[32768];   // 256 KB
  float4* A4 = (float4*)A;
  const int t  = threadIdx.x;
  const int s  = blockIdx.x >> 1;
  const int hb = blockIdx.x & 1;
  float2* own  = state + (size_t)s*65536 + (size_t)hb*32768;
  float4* own4 = (float4*)own;
  const float* gt = gtab + (size_t)s*NG*8;

  if (phase == 0) {                 // analytic init: |0...0>
    float4 z = make_float4(0.f, 0.f, 0.f, 0.f);
    for (int k = 0; k < 16; ++k) A4[t + 1024*k] = z;
    if (hb == 0 && t == 0) A[0] = make_float2(1.f, 0.f);
  } else {                          // async bulk-stage global -> LDS
    for (int k = 0; k < 16; ++k) { int i = t + 1024*k; async_g2l_b128(own4 + i, A4 + i); }
    wait_async0();
  }
  __syncthreads();

  int gbase;
  if (phase) {                      // CRX(15,0) of layer 1: control bit15 == hb
    if (hb) {
      cplx U[4]; loadmat(gt + 29*8, U);
      for (int k = 0; k < 16; ++k) {
        int j = t + 1024*k;         // target wire 0: pairs (2j, 2j+1)
        float2 a = A[2*j], b = A[2*j + 1];
        mixp(a, b, U);
        A[2*j] = a; A[2*j + 1] = b;
      }
    }
    __syncthreads();
    gbase = 30;
  } else {
    gbase = 0;
  }

  // fused 1q chain, wires 0..14: radix-4 quads for (w, w+1), then single w=14
  const float* g1 = gt + (size_t)gbase*8;
  for (int w = 0; w < 14; w += 2) {
    cplx Ua[4], Ub[4];
    loadmat(g1 + (size_t)w*8, Ua);
    loadmat(g1 + (size_t)(w+1)*8, Ub);
    int lmask = (1 << w) - 1;
    for (int k = 0; k < 8; ++k) {   // 8192 quads
      int j = t + 1024*k;
      int i00 = ((j >> w) << (w + 2)) | (j & lmask);
      int i01 = i00 | (1 << w);
      int i10 = i00 | (2 << w);
      int i11 = i10 | (1 << w);
      float2 a = A[i00], b = A[i01], c = A[i10], d = A[i11];
      mixp(a, b, Ua); mixp(c, d, Ua);    // gate on wire w
      mixp(a, c, Ub); mixp(b, d, Ub);    // gate on wire w+1
      A[i00] = a; A[i01] = b; A[i10] = c; A[i11] = d;
    }
    __syncthreads();
  }
  {
    cplx U[4]; loadmat(g1 + 14*8, U);
    for (int k = 0; k < 16; ++k) {  // wire 14: pairs (j, j+16384)
      int j = t + 1024*k;
      float2 a = A[j], b = A[j + 16384];
      mixp(a, b, U);
      A[j] = a; A[j + 16384] = b;
    }
    __syncthreads();
  }

  // CRX ring gates i -> i+1 for i = 0..13 (control bit i, target bit i+1)
  for (int gg = 0; gg < 14; ++gg) {
    int wt = gg + 1, wc = gg;
    cplx U[4]; loadmat(gt + (size_t)(gbase + 15 + gg)*8, U);
    int lmask = (1 << wt) - 1;
    for (int k = 0; k < 16; ++k) {
      int j = t + 1024*k;
      int i0 = ((j >> wt) << (wt + 1)) | (j & lmask);
      if (!((i0 >> wc) & 1)) continue;
      int i1 = i0 | (1 << wt);
      float2 a = A[i0], b = A[i1];
      mixp(a, b, U);
      A[i0] = a; A[i1] = b;
    }
    __syncthreads();
  }

  // async bulk-stage LDS -> global
  for (int k = 0; k < 16; ++k) { int i = t + 1024*k; async_l2g_b128(own4 + i, A4 + i); }
  wait_async0();
}

// ---------------- cross pass: composed [1q(15); CRX(14,15)] on pairs (i, i+2^15) ----------------
__global__ __launch_bounds__(256) void cross_pass(float2* __restrict__ state,
                                                  const float* __restrict__ xmatL)
{
  size_t gid = (size_t)blockIdx.x * 256 + threadIdx.x;  // one float4-pair each
  int s  = (int)(gid >> 14);                            // 16384 float4 per half
  int i4 = (int)(gid & 16383);
  float4* h0 = (float4*)(state + (size_t)s*65536);
  float4* h1 = (float4*)(state + (size_t)s*65536 + 32768);
  const float* Ms = xmatL + (size_t)s*16 + (((i4 >> 13) & 1) ? 8 : 0);  // bit14 select
  cplx m00{Ms[0],Ms[1]}, m01{Ms[2],Ms[3]}, m10{Ms[4],Ms[5]}, m11{Ms[6],Ms[7]};
  float4 f0 = h0[i4], f1 = h1[i4];
  mix2(f0.x, f0.y, f1.x, f1.y, m00, m01, m10, m11);
  mix2(f0.z, f0.w, f1.z, f1.w, m00, m01, m10, m11);
  h0[i4] = f0; h1[i4] = f1;
}

// ---------------- final pass: CRX(15,0)@L2 + <Z_w> expvals -> feats ----------------
__global__ __launch_bounds__(1024) void final_pass(const float2* __restrict__ state,
                                                   const float* __restrict__ xtra,
                                                   float* __restrict__ feats)
{
  __shared__ float wb[512];
  const int t = threadIdx.x, s = blockIdx.x;
  const float4* h0 = (const float4*)(state + (size_t)s*65536);
  const float4* h1 = (const float4*)(state + (size_t)s*65536 + 32768);
  float c15 = xtra[0], s15 = xtra[1];
  cplx U0{c15, 0.f}, U1{0.f, -s15}, U2{0.f, -s15}, U3{c15, 0.f};

  float acc[16];
  #pragma unroll
  for (int w = 0; w < 16; ++w) acc[w] = 0.f;

  for (int k = 0; k < 16; ++k) {
    int ip = t + 1024*k;                 // wire-0 pair index within a half
    float4 f0 = h0[ip];                  // half bit15 = 0 (no gate)
    float p0 = f0.x*f0.x + f0.y*f0.y;
    float p1 = f0.z*f0.z + f0.w*f0.w;
    float ps = p0 + p1;
    acc[0] += p0 - p1;
    #pragma unroll
    for (int w = 1; w < 15; ++w) acc[w] += ((ip >> (w-1)) & 1) ? -ps : ps;
    acc[15] += ps;

    float4 f1 = h1[ip];                  // half bit15 = 1: apply RX(crx2[15]) on wire 0
    mix2(f1.x, f1.y, f1.z, f1.w, U0, U1, U2, U3);
    float q0 = f1.x*f1.x + f1.y*f1.y;
    float q1 = f1.z*f1.z + f1.w*f1.w;
    float qs = q0 + q1;
    acc[0] += q0 - q1;
    #pragma unroll
    for (int w = 1; w < 15; ++w) acc[w] += ((ip >> (w-1)) & 1) ? -qs : qs;
    acc[15] -= qs;
  }

  #pragma unroll
  for (int w = 0; w < 16; ++w) {
    #pragma unroll
    for (int off = 16; off >= 1; off >>= 1) acc[w] += __shfl_xor(acc[w], off);
  }
  if ((t & 31) == 0) {
    #pragma unroll
    for (int w = 0; w < 16; ++w) wb[(t >> 5)*16 + w] = acc[w];
  }
  __syncthreads();
  if (t < NQ) {
    float f = 0.f;
    for (int wv = 0; wv < 32; ++wv) f += wb[wv*16 + t];
    feats[s*NQ + t] = f;
  }
}

// ---- head: logits = feats @ W^T + b, fused log_softmax, via V_WMMA_F32_16X16X4_F32 ----
typedef float v2f __attribute__((ext_vector_type(2)));
typedef float v8f __attribute__((ext_vector_type(8)));
#define NTILES 7   // ceil(100/16)

__global__ __launch_bounds__(32) void head_kernel(
    const float* __restrict__ feats, const float* __restrict__ Wm,
    const float* __restrict__ bias,  float* __restrict__ out)
{
  const int lane = threadIdx.x;       // one wave32 per block
  const int mt   = blockIdx.x;        // 16-row tile of samples
  const int g    = lane >> 4;
  const int l    = lane & 15;

  v8f acc[NTILES];
  #pragma unroll
  for (int nt = 0; nt < NTILES; ++nt) acc[nt] = {};

  #pragma unroll
  for (int k0 = 0; k0 < NQ; k0 += 4) {
    v2f a;
    a.x = feats[(mt*16 + l)*NQ + k0 + 2*g + 0];
    a.y = feats[(mt*16 + l)*NQ + k0 + 2*g + 1];
    #pragma unroll
    for (int nt = 0; nt < NTILES; ++nt) {
      int n  = nt*16 + l;
      int nc = n < NCLS ? n : (NCLS - 1);
      v2f b;
      b.x = Wm[nc*NQ + k0 + 2*g + 0];
      b.y = Wm[nc*NQ + k0 + 2*g + 1];
      b.x = (n < NCLS) ? b.x : 0.f;
      b.y = (n < NCLS) ? b.y : 0.f;
      acc[nt] = __builtin_amdgcn_wmma_f32_16x16x4_f32(
          false, a, false, b, (short)0, acc[nt], false, false);
    }
  }

  #pragma unroll
  for (int r = 0; r < 8; ++r) {       // row M = r + 8g, N = nt*16 + l
    float lg[NTILES];
    #pragma unroll
    for (int nt = 0; nt < NTILES; ++nt) {
      int n = nt*16 + l;
      lg[nt] = (n < NCLS) ? (acc[nt][r] + bias[n < NCLS ? n : 0]) : -3.0e38f;
    }
    float m = lg[0];
    #pragma unroll
    for (int nt = 1; nt < NTILES; ++nt) m = fmaxf(m, lg[nt]);
    #pragma unroll
    for (int off = 8; off >= 1; off >>= 1) m = fmaxf(m, __shfl_xor(m, off));
    float ssum = 0.f;
    #pragma unroll
    for (int nt = 0; nt < NTILES; ++nt) {
      int n = nt*16 + l;
      if (n < NCLS) ssum += expf(lg[nt] - m);
    }
    #pragma unroll
    for (int off = 8; off >= 1; off >>= 1) ssum += __shfl_xor(ssum, off);
    float lse = m + logf(ssum);
    int row = mt*16 + r + 8*g;
    #pragma unroll
    for (int nt = 0; nt < NTILES; ++nt) {
      int n = nt*16 + l;
      if (n < NCLS) out[row*NCLS + n] = lg[nt] - lse;
    }
  }
}

extern "C" void kernel_launch(void* const* d_in, const int* in_sizes, int n_in,
                              void* d_out, int out_size, void* d_ws, size_t ws_size,
                              hipStream_t stream) {
  (void)in_sizes; (void)n_in; (void)out_size; (void)ws_size;
  const float* x    = (const float*)d_in[0];
  const float* rot1 = (const float*)d_in[1];
  const float* crx1 = (const float*)d_in[2];
  const float* rot2 = (const float*)d_in[3];
  const float* crx2 = (const float*)d_in[4];
  const float* Wm   = (const float*)d_in[5];
  const float* b    = (const float*)d_in[6];

  float*  ws    = (float*)d_ws;
  float*  feats = ws + WS_FEATS;
  float*  gtab  = ws + WS_GTAB;
  float*  xmat  = ws + WS_XMAT;
  float*  xtra  = ws + WS_XTRA;
  float2* stateP = (float2*)(ws + WS_STATE);     // 512 samples x 65536 complex

  prep_kernel<<<(NBATCH*64)/256, 256, 0, stream>>>(x, rot1, crx1, rot2, crx2, gtab, xmat, xtra);
  local_pass<<<2*NBATCH, 1024, 0, stream>>>(stateP, gtab, 0);                       // L1 local
  cross_pass<<<(NBATCH*16384)/256, 256, 0, stream>>>(stateP, xmat);                 // L1 wire-15
  local_pass<<<2*NBATCH, 1024, 0, stream>>>(stateP, gtab, 1);                       // L2 local
  cross_pass<<<(NBATCH*16384)/256, 256, 0, stream>>>(stateP, xmat + (size_t)NBATCH*16); // L2 wire-15
  final_pass<<<NBATCH, 1024, 0, stream>>>(stateP, xtra, feats);                     // expvals
  head_kernel<<<NBATCH/16, 32, 0, stream>>>(feats, Wm, b, (float*)d_out);           // WMMA head
}